// BoundaryKDV6_24979529793878
// MI455X (gfx1250) — compile-verified
//
#include <hip/hip_runtime.h>
#include <math.h>

// Problem constants (from reference)
#define B_   2
#define C_   14
#define H_   128
#define W_   128
#define HW   (H_*W_)          // 16384
#define NP   (B_*(C_-1))      // 26 (batch,class) pairs
#define NIMG (2*NP)           // 52 edge-images (pred + targ per pair)

typedef float v2f __attribute__((ext_vector_type(2)));
typedef float v8f __attribute__((ext_vector_type(8)));

// ---------------------------------------------------------------------------
// K1: per-pixel argmax over C for student & teacher; also zero validity flags.
// ---------------------------------------------------------------------------
__global__ void bkd_argmax(const float* __restrict__ ls, const float* __restrict__ lt,
                           int* __restrict__ ps, int* __restrict__ pt,
                           int* __restrict__ flags) {
    int t = blockIdx.x * blockDim.x + threadIdx.x;
    if (t < NIMG * 2) flags[t] = 0;          // [img][{any_fg, any_bg}]
    if (t >= B_ * HW) return;
    int b = t / HW, p = t % HW;
    const float* s0 = ls + (size_t)b * C_ * HW + p;
    const float* t0 = lt + (size_t)b * C_ * HW + p;
    float bs = s0[0]; int as = 0;            // first-max tie rule (strict >)
    float bt = t0[0]; int at = 0;
    for (int c = 1; c < C_; ++c) {
        float vs = s0[c * HW]; if (vs > bs) { bs = vs; as = c; }
        float vt = t0[c * HW]; if (vt > bt) { bt = vt; at = c; }
    }
    ps[t] = as; pt[t] = at;
}

// ---------------------------------------------------------------------------
// K2: build pred/targ edge images per pair; record any_fg / any_bg flags.
// edge = mask & ~erosion(mask)  (cross element, borders erode)
// ---------------------------------------------------------------------------
__global__ void bkd_images(const float* __restrict__ ls,
                           const int* __restrict__ ps, const int* __restrict__ pt,
                           float* __restrict__ pimg, float* __restrict__ timg,
                           int* __restrict__ flags) {
    int t = blockIdx.x * blockDim.x + threadIdx.x;
    if (t >= NP * HW) return;
    int n = t / HW, p = t % HW;
    int i = p >> 7, j = p & 127;
    int b = n / (C_ - 1), c = n % (C_ - 1) + 1;
    const int* psb = ps + b * HW;
    const int* ptb = pt + b * HW;

    bool m  = psb[p] == c;
    bool er = m && (i > 0      && psb[p - W_] == c)
                && (i < H_ - 1 && psb[p + W_] == c)
                && (j > 0      && psb[p - 1]  == c)
                && (j < W_ - 1 && psb[p + 1]  == c);
    float pred = (m && !er) ? ls[((size_t)b * C_ + c) * HW + p] : 0.0f;

    bool mt  = ptb[p] == c;
    bool ert = mt && (i > 0      && ptb[p - W_] == c)
                  && (i < H_ - 1 && ptb[p + W_] == c)
                  && (j > 0      && ptb[p - 1]  == c)
                  && (j < W_ - 1 && ptb[p + 1]  == c);
    float targ = (mt && !ert) ? (float)ptb[p] : 0.0f;

    pimg[n * HW + p] = pred;
    timg[n * HW + p] = targ;

    // validity flags (idempotent OR -> deterministic)
    if (pred > 0.5f) atomicOr(&flags[2 * n + 0], 1);
    else             atomicOr(&flags[2 * n + 1], 1);
    if (targ > 0.5f) atomicOr(&flags[2 * (NP + n) + 0], 1);
    else             atomicOr(&flags[2 * (NP + n) + 1], 1);
}

// ---------------------------------------------------------------------------
// K3: phase-1 EDT: exact per-column 1-D distance to nearest mask pixel for
// both masks (~fg and fg) of every image. u8 output; 255 = "empty column"
// sentinel (never wins the phase-2 min when any column is non-empty; the
// all-empty case is zeroed by the validity flags, matching the reference).
// One thread per (image, mask, column); fg bits cached in 4 registers.
// ---------------------------------------------------------------------------
__global__ void bkd_edt1(const float* __restrict__ pimg, const float* __restrict__ timg,
                         unsigned char* __restrict__ d1) {
    int t = blockIdx.x * blockDim.x + threadIdx.x;
    if (t >= NIMG * 2 * W_) return;
    int img  = t >> 8;            // 2 masks * 128 cols = 256 threads per image
    int r    = t & 255;
    int mask = r >> 7;            // 0: ~fg, 1: fg
    int q    = r & 127;
    const float* src = (img < NP) ? (pimg + img * HW) : (timg + (img - NP) * HW);

    unsigned int bits[4] = {0u, 0u, 0u, 0u};
    for (int i = 0; i < H_; ++i) {
        bool fg = src[i * W_ + q] > 0.5f;
        bool mb = mask ? fg : !fg;
        if (mb) bits[i >> 5] |= (1u << (i & 31));
    }
    unsigned char* out = d1 + ((size_t)(img * 2 + mask)) * HW;
    int d = 255;
    for (int i = H_ - 1; i >= 0; --i) {       // backward scan
        bool mb = (bits[i >> 5] >> (i & 31)) & 1u;
        d = mb ? 0 : min(d + 1, 255);
        out[i * W_ + q] = (unsigned char)d;
    }
    d = 255;
    for (int i = 0; i < H_; ++i) {            // forward scan + combine
        bool mb = (bits[i >> 5] >> (i & 31)) & 1u;
        d = mb ? 0 : min(d + 1, 255);
        int prev = out[i * W_ + q];
        out[i * W_ + q] = (unsigned char)min(prev, d);
    }
}

// ---------------------------------------------------------------------------
// K4: phase-2 min-plus + weighted row sum. One block per (pair, row).
// d1^2 rows staged in LDS and broadcast to all 128 lanes of the min-plus.
// Row reduction done with V_WMMA_F32_16X16X4_F32: D = A(16x4 data) x B(4x16
// ones) + C accumulates exact row sums deterministically inside wave 0.
// A layout per ISA: lane<16 holds A[m][0..1], lane>=16 holds A[m][2..3].
// ---------------------------------------------------------------------------
__global__ void bkd_rowfield(const float* __restrict__ pimg, const float* __restrict__ timg,
                             const unsigned char* __restrict__ d1,
                             const int* __restrict__ flags,
                             float* __restrict__ rowsum) {
    __shared__ float pa[W_], pb[W_], ta[W_], tb[W_], warr[W_];
    int n = blockIdx.x, i = blockIdx.y, j = threadIdx.x;
    int imgp = n, imgt = NP + n;
    {
        int vp0 = d1[((size_t)(imgp * 2 + 0)) * HW + i * W_ + j];
        int vp1 = d1[((size_t)(imgp * 2 + 1)) * HW + i * W_ + j];
        int vt0 = d1[((size_t)(imgt * 2 + 0)) * HW + i * W_ + j];
        int vt1 = d1[((size_t)(imgt * 2 + 1)) * HW + i * W_ + j];
        pa[j] = (float)(vp0 * vp0);
        pb[j] = (float)(vp1 * vp1);
        ta[j] = (float)(vt0 * vt0);
        tb[j] = (float)(vt1 * vt1);
    }
    float predv = pimg[n * HW + i * W_ + j];
    float targv = timg[n * HW + i * W_ + j];
    bool vp = flags[2 * imgp + 0] && flags[2 * imgp + 1];
    bool vt = flags[2 * imgt + 0] && flags[2 * imgt + 1];
    __syncthreads();

    float m1 = 3.4e38f, m2 = m1, m3 = m1, m4 = m1;
    for (int q = 0; q < W_; ++q) {
        float off = (float)((q - j) * (q - j));
        m1 = fminf(m1, pa[q] + off);
        m2 = fminf(m2, pb[q] + off);
        m3 = fminf(m3, ta[q] + off);
        m4 = fminf(m4, tb[q] + off);
    }
    float f = (vp ? (m1 + m2) : 0.0f) + (vt ? (m3 + m4) : 0.0f);
    float diff = predv - targv;
    warr[j] = diff * diff * f;
    __syncthreads();

    if (j < 32) {                               // wave 0: EXEC all ones
        v8f c = {0.f, 0.f, 0.f, 0.f, 0.f, 0.f, 0.f, 0.f};
        v2f bones; bones.x = 1.0f; bones.y = 1.0f;   // B = ones(4x16)
        int mrow = j & 15, half = j >> 4;
        #pragma unroll
        for (int tile = 0; tile < 2; ++tile) {
            v2f a;                               // A[m][k] = warr[tile*64 + 4m + k]
            a.x = warr[tile * 64 + 4 * mrow + 2 * half];
            a.y = warr[tile * 64 + 4 * mrow + 2 * half + 1];
            c = __builtin_amdgcn_wmma_f32_16x16x4_f32(false, a, false, bones,
                                                      (short)0, c, false, false);
        }
        // per-lane: sum of 8 D rows of one half; combine halves across lane^16
        float s = c[0] + c[1] + c[2] + c[3] + c[4] + c[5] + c[6] + c[7];
        s += __shfl_xor(s, 16, 32);
        if (j == 0) rowsum[n * H_ + i] = s;
    }
}

// ---------------------------------------------------------------------------
// K5: deterministic finalize: hd = mean, loss = log(hd+1), out = 0.5*sum.
// ---------------------------------------------------------------------------
__global__ void bkd_final(const float* __restrict__ rowsum, float* __restrict__ out) {
    __shared__ float red[NP];
    int t = threadIdx.x;
    if (t < NP) {
        float s = 0.f;
        for (int i = 0; i < H_; ++i) s += rowsum[t * H_ + i];
        float hd = s * (1.0f / 16384.0f);
        red[t] = logf(hd + 1.0f);
    }
    __syncthreads();
    if (t == 0) {
        float tot = 0.f;
        for (int k = 0; k < NP; ++k) tot += red[k];
        out[0] = 0.5f * tot;                  // (1 - loss_weight) * sum
    }
}

// ---------------------------------------------------------------------------
extern "C" void kernel_launch(void* const* d_in, const int* in_sizes, int n_in,
                              void* d_out, int out_size, void* d_ws, size_t ws_size,
                              hipStream_t stream) {
    const float* ls = (const float*)d_in[0];   // [B,C,H,W] f32
    const float* lt = (const float*)d_in[1];   // [B,C,H,W] f32
    float* out = (float*)d_out;                // scalar f32

    // workspace layout (~5.4 MB)
    int*   ps     = (int*)d_ws;                              // B*HW ints
    int*   pt     = ps + B_ * HW;                            // B*HW ints
    float* pimg   = (float*)(pt + B_ * HW);                  // NP*HW f32
    float* timg   = pimg + (size_t)NP * HW;                  // NP*HW f32
    int*   flags  = (int*)(timg + (size_t)NP * HW);          // NIMG*2 ints
    unsigned char* d1 = (unsigned char*)(flags + NIMG * 2);  // NIMG*2*HW u8
    float* rowsum = (float*)(d1 + (size_t)NIMG * 2 * HW);    // NP*H f32

    (void)in_sizes; (void)n_in; (void)out_size; (void)ws_size;

    bkd_argmax  <<<(B_ * HW + 255) / 256, 256, 0, stream>>>(ls, lt, ps, pt, flags);
    bkd_images  <<<(NP * HW + 255) / 256, 256, 0, stream>>>(ls, ps, pt, pimg, timg, flags);
    bkd_edt1    <<<(NIMG * 2 * W_ + 255) / 256, 256, 0, stream>>>(pimg, timg, d1);
    bkd_rowfield<<<dim3(NP, H_), 128, 0, stream>>>(pimg, timg, d1, flags, rowsum);
    bkd_final   <<<1, 32, 0, stream>>>(rowsum, out);
}